// RegionLoss_NoClass_1Bbox_80023830659722
// MI455X (gfx1250) — compile-verified
//
#include <hip/hip_runtime.h>
#include <hip/hip_bf16.h>

#define NAC 5
#define HW 76
#define CHS (HW * HW)       /* 5776  : channel stride                */
#define PERB (NAC * CHS)    /* 28880 : cells per batch (b,a,cell)    */
#define BS 512
#define TOTAL (BS * PERB)   /* 14,786,560 cells                      */
#define GROUPS (TOTAL / 4)  /* 3,696,640 float4 groups               */
#define PERB4 (PERB / 4)    /* 7220                                  */
#define CHS4 (CHS / 4)      /* 1444                                  */
#define OBJECT_SCALE 5.0f
#define NO_OBJECT_SCALE 1.0f
#define SIL_THRESH 0.6f

__constant__ float c_aw[NAC] = {0.57273f, 1.87446f, 3.33843f, 7.88282f, 9.77052f};
__constant__ float c_ah[NAC] = {0.677385f, 2.06253f, 5.47434f, 3.52778f, 9.16828f};

typedef __attribute__((ext_vector_type(2))) float v2f;
typedef __attribute__((ext_vector_type(8))) float v8f;

__device__ __forceinline__ float sigmoidf_(float x) {
    return 1.0f / (1.0f + __expf(-x));
}

__device__ __forceinline__ float bbox_iou_(float x1, float y1, float w1, float h1,
                                           float x2, float y2, float w2, float h2) {
    float mx = fminf(x1 - w1 * 0.5f, x2 - w2 * 0.5f);
    float Mx = fmaxf(x1 + w1 * 0.5f, x2 + w2 * 0.5f);
    float my = fminf(y1 - h1 * 0.5f, y2 - h2 * 0.5f);
    float My = fmaxf(y1 + h1 * 0.5f, y2 + h2 * 0.5f);
    float cw = w1 + w2 - (Mx - mx);
    float ch = h1 + h2 - (My - my);
    float carea = (cw <= 0.0f || ch <= 0.0f) ? 0.0f : cw * ch;
    float uarea = w1 * h1 + w2 * h2 - carea;
    return carea / uarea;
}

// ---------------------------------------------------------------------------
// Kernel 1: per-batch prep.  ws[0..15] = accumulators (zeroed here).
// ws[16 + b*16 .. +15] = {gx, gy, gw, gh, soff(int), tx_s, ty_s, tw_s, th_s,
//                         iou_t, 0...}
// ---------------------------------------------------------------------------
__global__ void region_prep_kernel(const float* __restrict__ pred,
                                   const float* __restrict__ target,
                                   float* __restrict__ ws) {
    int b = blockIdx.x * blockDim.x + threadIdx.x;
    if (b < 16) ws[b] = 0.0f;
    if (b >= BS) return;

    float gx = target[b * 4 + 0] * (float)HW;
    float gy = target[b * 4 + 1] * (float)HW;
    float gw = target[b * 4 + 2] * (float)HW;
    float gh = target[b * 4 + 3] * (float)HW;

    int best = 0;
    float bestiou = -1.0f;
#pragma unroll
    for (int a = 0; a < NAC; ++a) {
        float iou = bbox_iou_(0.0f, 0.0f, gw, gh, 0.0f, 0.0f, c_aw[a], c_ah[a]);
        if (iou > bestiou) { bestiou = iou; best = a; }
    }

    int gi = (int)gx;
    int gj = (int)gy;
    int cell = gj * HW + gi;

    size_t base = (size_t)(b * 25 + best * 5) * CHS + (size_t)cell;
    float p0 = pred[base];
    float p1 = pred[base + CHS];
    float p2 = pred[base + 2 * CHS];
    float p3 = pred[base + 3 * CHS];

    float bx = sigmoidf_(p0) + (float)gi;
    float by = sigmoidf_(p1) + (float)gj;
    float bw = __expf(p2) * c_aw[best];
    float bh = __expf(p3) * c_ah[best];
    float iou_t = bbox_iou_(gx, gy, gw, gh, bx, by, bw, bh);

    float* row = ws + 16 + b * 16;
    row[0] = gx;
    row[1] = gy;
    row[2] = gw;
    row[3] = gh;
    row[4] = __int_as_float(best * CHS + cell);  // batch-local special offset
    row[5] = gx - (float)gi;
    row[6] = gy - (float)gj;
    row[7] = __logf(gw / c_aw[best]);
    row[8] = __logf(gh / c_ah[best]);
    row[9] = iou_t;
    row[10] = 0.0f; row[11] = 0.0f; row[12] = 0.0f;
    row[13] = 0.0f; row[14] = 0.0f; row[15] = 0.0f;
}

// ---------------------------------------------------------------------------
// Kernel 2: streaming reduction; 4 cells (one float4 per channel) per thread
// per grid-stride step.  Block epilogue: exact f32 block reduction via
// chained V_WMMA_F32_16X16X4_F32 with A == ones(16x4):
//   D[m,n] = sum_k B[k,n] + C[m,n]   (column sums of the LDS partials).
// ---------------------------------------------------------------------------
__global__ void __launch_bounds__(256)
region_main_kernel(const float* __restrict__ pred,
                   float* __restrict__ ws) {
    const float* table = ws + 16;
    float acc0 = 0.0f, acc1 = 0.0f, acc2 = 0.0f, acc3 = 0.0f, acc4 = 0.0f;

    int stride = gridDim.x * blockDim.x;
    for (int g = blockIdx.x * blockDim.x + threadIdx.x; g < GROUPS; g += stride) {
        int b    = g / PERB4;
        int r4   = g - b * PERB4;          // a*CHS4 + cellgroup
        int a    = r4 / CHS4;
        int cg   = r4 - a * CHS4;
        int cell = cg * 4;                 // j*HW + i, multiple of 4
        int jj   = cell / HW;
        int ii   = cell - jj * HW;
        int rbase = a * CHS + cell;        // batch-local flat cell index

        const float4 t0 = *(const float4*)(table + b * 16);      // gx gy gw gh
        const float4 t1 = *(const float4*)(table + b * 16 + 4);  // soff tx ty tw
        const float4 t2 = *(const float4*)(table + b * 16 + 8);  // th iou_t . .
        int soff = __float_as_int(t1.x);

        size_t base = (size_t)(b * 25 + a * 5) * CHS + (size_t)cell;
        float4 q0 = *(const float4*)(pred + base);            // x logits
        float4 q1 = *(const float4*)(pred + base + CHS);      // y logits
        float4 q2 = *(const float4*)(pred + base + 2 * CHS);  // w raw
        float4 q3 = *(const float4*)(pred + base + 3 * CHS);  // h raw
        float4 q4 = *(const float4*)(pred + base + 4 * CHS);  // conf logits

        // Speculative prefetch one grid-stride ahead of this stream
        // (gfx1250 global_prefetch_b8; bad tail addresses silently dropped).
        __builtin_prefetch(pred + base + (size_t)stride * 20, 0, 1);

        float aw = c_aw[a], ah = c_ah[a];

#pragma unroll
        for (int k = 0; k < 4; ++k) {
            float p0 = ((const float*)&q0)[k];
            float p1 = ((const float*)&q1)[k];
            float p2 = ((const float*)&q2)[k];
            float p3 = ((const float*)&q3)[k];
            float p4 = ((const float*)&q4)[k];

            float sx = sigmoidf_(p0);
            float sy = sigmoidf_(p1);
            float pconf = sigmoidf_(p4);
            float bx = sx + (float)ii;
            float by = sy + (float)jj;
            float bw = __expf(p2) * aw;
            float bh = __expf(p3) * ah;

            float iou_all = bbox_iou_(t0.x, t0.y, t0.z, t0.w, bx, by, bw, bh);

            bool special = ((rbase + k) == soff);
            float cmask = special ? OBJECT_SCALE
                                  : (iou_all > SIL_THRESH ? 0.0f : NO_OBJECT_SCALE);
            float tbx = special ? t1.y : 0.5f;
            float tby = special ? t1.z : 0.5f;
            float tbw = special ? t1.w : 0.0f;
            float tbh = special ? t2.x : 0.0f;
            float tcf = special ? t2.y : 0.0f;

            float dx = sx - tbx;    acc0 += dx * dx;
            float dy = sy - tby;    acc1 += dy * dy;
            float dw = p2 - tbw;    acc2 += dw * dw;
            float dh = p3 - tbh;    acc3 += dh * dh;
            float dc = pconf - tcf; acc4 += cmask * dc * dc;

            // branchless row wrap (HW == 76 is a multiple of 4, so no wrap within a group)
            ++ii;
            int wrap = (ii == HW);
            jj += wrap;
            ii = wrap ? 0 : ii;
        }
    }

    // ---- block reduction: LDS stage, then wave0 WMMA column-sum chain ----
    __shared__ float part[256][16];
    int tid = threadIdx.x;
    part[tid][0] = acc0;
    part[tid][1] = acc1;
    part[tid][2] = acc2;
    part[tid][3] = acc3;
    part[tid][4] = acc4;
#pragma unroll
    for (int c = 5; c < 16; ++c) part[tid][c] = 0.0f;
    __syncthreads();

    if (tid >= 32) return;   // wave 0 only: EXEC all-ones for WMMA

    v2f av;  av.x = 1.0f;  av.y = 1.0f;         // A = ones(16x4)
    v8f cv = {0.0f, 0.0f, 0.0f, 0.0f, 0.0f, 0.0f, 0.0f, 0.0f};

    int n  = tid & 15;              // B column (N = lane % 16)
    int kg = (tid >> 4) << 1;       // K group: lanes 0-15 -> K0/K1, 16-31 -> K2/K3

    for (int t = 0; t < 256; t += 4) {
        v2f bv;
        bv.x = part[t + kg][n];
        bv.y = part[t + kg + 1][n];
        // D[m,n] = sum_k A[m,k]*B[k,n] + C[m,n]  ->  column sums of part
        cv = __builtin_amdgcn_wmma_f32_16x16x4_f32(
            /*neg_a=*/false, av, /*neg_b=*/false, bv,
            /*c_mod=*/(short)0, cv, /*reuse_a=*/false, /*reuse_b=*/false);
    }

    // D[0][n] lives in VGPR 0 of lane n (n = 0..15); components are n = 0..4
    if (tid < 5) atomicAdd(&ws[tid], cv[0]);
}

// ---------------------------------------------------------------------------
// Kernel 3: finalize the six scalars.
// ---------------------------------------------------------------------------
__global__ void region_finalize_kernel(const float* __restrict__ ws,
                                       float* __restrict__ out) {
    if (threadIdx.x == 0 && blockIdx.x == 0) {
        float lx = ws[0] * 0.5f;
        float ly = ws[1] * 0.5f;
        float lw = ws[2] * 0.5f;
        float lh = ws[3] * 0.5f;
        float lc = ws[4] * 0.5f;
        out[0] = lx + ly + lw + lh + lc;
        out[1] = lx;
        out[2] = ly;
        out[3] = lw;
        out[4] = lh;
        out[5] = lc;
    }
}

extern "C" void kernel_launch(void* const* d_in, const int* in_sizes, int n_in,
                              void* d_out, int out_size, void* d_ws, size_t ws_size,
                              hipStream_t stream) {
    const float* pred   = (const float*)d_in[0];
    const float* target = (const float*)d_in[1];
    float* out = (float*)d_out;
    float* ws  = (float*)d_ws;   // needs (16 + 512*16)*4 = 32,832 bytes

    region_prep_kernel<<<2, 256, 0, stream>>>(pred, target, ws);
    region_main_kernel<<<2048, 256, 0, stream>>>(pred, ws);
    region_finalize_kernel<<<1, 32, 0, stream>>>(ws, out);
}